// YoloPredict_13030930776077
// MI455X (gfx1250) — compile-verified
//
#include <hip/hip_runtime.h>
#include <stdint.h>

#define NCLS   20
#define NANCH  5
#define HF     19
#define WF     19
#define HW     (HF*WF)           // 361
#define NBOX   (NANCH*HW)        // 1805
#define NWORDS 57                // ceil(1805/32)
#define WPAD   64                // padded words per bitmask row (256B)
#define BATCH  16
#define NMS_THF 0.45f

__device__ const float BIAS_DEV[10] = {1.08f, 1.19f, 3.42f, 4.41f, 6.63f,
                                       11.38f, 9.42f, 5.11f, 16.62f, 10.52f};

typedef unsigned int v4u __attribute__((ext_vector_type(4)));
typedef int          v8i __attribute__((ext_vector_type(8)));
typedef int          v4i __attribute__((ext_vector_type(4)));

// ---------------------------------------------------------------- decode ----
__global__ void __launch_bounds__(256)
decode_kernel(const float* __restrict__ x, const float* __restrict__ im_info,
              float* __restrict__ boxes, float* __restrict__ scores) {
  int n = blockIdx.x * blockDim.x + threadIdx.x;
  int b = blockIdx.y;
  if (n >= NBOX) return;
  int a = n / HW;
  int r = n - a * HW;
  int h = r / WF;
  int w = r - h * WF;

  size_t base = (size_t)b * 125 * HW + (size_t)h * WF + w;

  float tx = x[base + (size_t)(a * 2 + 0) * HW];
  float ty = x[base + (size_t)(a * 2 + 1) * HW];
  float tw = x[base + (size_t)(10 + a * 2) * HW];
  float th = x[base + (size_t)(11 + a * 2) * HW];
  float to = x[base + (size_t)(20 + a) * HW];

  float sx = 1.0f / (1.0f + expf(-tx));
  float sy = 1.0f / (1.0f + expf(-ty));
  float so = 1.0f / (1.0f + expf(-to));

  float im_h = im_info[b * 2 + 0];
  float im_w = im_info[b * 2 + 1];

  float bx = (sx + (float)w) / (float)WF * im_w;
  float by = (sy + (float)h) / (float)HF * im_h;
  float bw = expf(tw) * BIAS_DEV[a * 2 + 0] / (float)WF * im_w;
  float bh = expf(th) * BIAS_DEV[a * 2 + 1] / (float)HF * im_h;

  float* bo = boxes + ((size_t)b * NBOX + n) * 4;
  bo[0] = bx; bo[1] = by; bo[2] = bw; bo[3] = bh;

  // softmax over 20 classes * obj  (match jax: exp(x-max)/sum, then *obj)
  float cv[NCLS];
  float m = -3.402823466e38f;
#pragma unroll
  for (int k = 0; k < NCLS; ++k) {
    cv[k] = x[base + (size_t)(25 + a * NCLS + k) * HW];
    m = fmaxf(m, cv[k]);
  }
  float ssum = 0.0f;
#pragma unroll
  for (int k = 0; k < NCLS; ++k) { cv[k] = expf(cv[k] - m); ssum += cv[k]; }
#pragma unroll
  for (int k = 0; k < NCLS; ++k)
    scores[((size_t)b * NCLS + k) * NBOX + n] = (cv[k] / ssum) * so;
}

// ------------------------------------------------------------- IoU bitmask --
__global__ void __launch_bounds__(256)
ioubits_kernel(const float* __restrict__ boxes, uint32_t* __restrict__ bits) {
  __shared__ float4 lb[NBOX];     // 28.9 KB, staged via TDM
  int b = blockIdx.y;
  const float* src = boxes + (size_t)b * NBOX * 4;

  if (threadIdx.x == 0) {
    // Build Tensor DMA descriptor: 1-D copy of NBOX*4 f32 elements -> LDS.
    uint32_t lds_off = (uint32_t)(uintptr_t)(&lb[0]);
    uint64_t ga      = (uint64_t)(uintptr_t)src;
    uint32_t nelem   = NBOX * 4;                 // 7220 x 4B
    v4u g0;
    g0.x = 1u;                                   // count = 1 valid descriptor
    g0.y = lds_off;                              // lds_addr
    g0.z = (uint32_t)(ga & 0xffffffffu);         // global_addr[31:0]
    g0.w = (uint32_t)((ga >> 32) & 0x01ffffffu)  // global_addr[56:32]
         | (2u << 30);                           // type = 2 ("image")
    v8i g1;
    g1[0] = (int)(2u << 16);                     // workgroup_mask=0, data_size=2 (4B)
    g1[1] = (int)((nelem & 0xffffu) << 16);      // tensor_dim0[15:0]
    g1[2] = (int)((nelem >> 16) | (1u << 16));   // tensor_dim0[31:16] | tensor_dim1=1
    g1[3] = (int)((nelem & 0xffffu) << 16);      // tensor_dim1 hi=0 | tile_dim0
    g1[4] = 0;                                   // tile_dim1=0, tile_dim2=0 (unused)
    g1[5] = (int)nelem;                          // tensor_dim0_stride[31:0]
    g1[6] = 0;
    g1[7] = 0;
    v4i gz = {0, 0, 0, 0};
    v8i gz8 = {0, 0, 0, 0, 0, 0, 0, 0};
    __builtin_amdgcn_tensor_load_to_lds(g0, g1, gz, gz, gz8, 0);
    __builtin_amdgcn_s_wait_tensorcnt(0);
  }
  __syncthreads();

  int g = blockIdx.x * 256 + threadIdx.x;        // word index within this batch
  if (g >= NBOX * NWORDS) return;
  int i  = g / NWORDS;
  int wv = g - i * NWORDS;

  float4 bi  = lb[i];
  float ix1 = bi.x - bi.z * 0.5f, ix2 = bi.x + bi.z * 0.5f;
  float iy1 = bi.y - bi.w * 0.5f, iy2 = bi.y + bi.w * 0.5f;
  float ai  = bi.z * bi.w;

  uint32_t bitsw = 0u;
  int j0 = wv << 5;
#pragma unroll 4
  for (int jj = 0; jj < 32; ++jj) {
    int j = j0 + jj;
    if (j < NBOX) {
      float4 bj = lb[j];
      float jx1 = bj.x - bj.z * 0.5f, jx2 = bj.x + bj.z * 0.5f;
      float jy1 = bj.y - bj.w * 0.5f, jy2 = bj.y + bj.w * 0.5f;
      float iw = fmaxf(fminf(ix2, jx2) - fmaxf(ix1, jx1), 0.0f);
      float ih = fmaxf(fminf(iy2, jy2) - fmaxf(iy1, jy1), 0.0f);
      float inter = iw * ih;
      float iou = inter / (ai + bj.z * bj.w - inter + 1e-9f);
      if (iou > NMS_THF && j != i) bitsw |= (1u << jj);   // diag cleared
    }
  }
  bits[((size_t)b * NBOX + i) * WPAD + wv] = bitsw;
}

// --------------------------------------------------------- sort + NMS scan --
__global__ void __launch_bounds__(256)
sortnms_kernel(const float* __restrict__ scores, const uint32_t* __restrict__ bits,
               float* __restrict__ prob) {
  __shared__ unsigned long long key[2048];   // packed (~mono(score))<<32 | idx
  __shared__ float sc[NBOX];
  int c = blockIdx.x, b = blockIdx.y;
  const float* s = scores + ((size_t)b * NCLS + c) * NBOX;

  for (int i = threadIdx.x; i < 2048; i += 256) {
    unsigned long long p;
    if (i < NBOX) {
      float f = s[i];
      sc[i] = f;
      uint32_t ub = __float_as_uint(f);
      uint32_t mono = (f >= 0.0f) ? (ub ^ 0x80000000u) : ~ub;  // order-preserving
      p = ((unsigned long long)(~mono) << 32) | (uint32_t)i;   // asc sort => desc score, asc idx
    } else {
      p = ~0ull;                                               // pad sorts last
    }
    key[i] = p;
  }
  __syncthreads();

  // bitonic sort ascending (2048 elems, 256 threads)
  for (int k = 2; k <= 2048; k <<= 1) {
    for (int j = k >> 1; j > 0; j >>= 1) {
      for (int i = threadIdx.x; i < 2048; i += 256) {
        int ixj = i ^ j;
        if (ixj > i) {
          unsigned long long a0 = key[i], a1 = key[ixj];
          bool up = ((i & k) == 0);
          if ((a0 > a1) == up) { key[i] = a1; key[ixj] = a0; }
        }
      }
      __syncthreads();
    }
  }

  // single-wave greedy scan; keep mask in 2 VGPRs/lane (64 words);
  // depth-8 register pipeline on bitmask rows (order-static addresses)
  if (threadIdx.x < 32) {
    const int D = 8;
    int lane = threadIdx.x;
    uint32_t keep0 = ~0u, keep1 = ~0u;
    const uint32_t* base = bits + (size_t)b * NBOX * WPAD;

    int      pidx[D];
    uint32_t pm0[D], pm1[D];
#pragma unroll
    for (int u = 0; u < D; ++u) {               // NBOX >> D, all valid
      pidx[u] = (int)(uint32_t)key[u];
      const uint32_t* r = base + (size_t)pidx[u] * WPAD;
      pm0[u] = r[lane * 2];
      pm1[u] = r[lane * 2 + 1];
    }

    for (int i = 0; i < NBOX; i += D) {
#pragma unroll
      for (int u = 0; u < D; ++u) {
        int ii = i + u;
        if (ii < NBOX) {
          int idx = pidx[u];
          uint32_t m0 = pm0[u], m1 = pm1[u];

          int nf = ii + D;                      // refill this pipeline slot
          if (nf < NBOX) {
            int fidx = (int)(uint32_t)key[nf];
            pidx[u] = fidx;
            const uint32_t* r = base + (size_t)fidx * WPAD;
            pm0[u] = r[lane * 2];
            pm1[u] = r[lane * 2 + 1];
          }

          int pf = ii + 2 * D;                  // warm caches further ahead
          if (pf < NBOX && (lane & 15) == 0) {  // one prefetch per 128B line
            int qidx = (int)(uint32_t)key[pf];
            __builtin_prefetch((const void*)(base + (size_t)qidx * WPAD + lane * 2), 0, 0);
          }

          int wv = idx >> 5;
          uint32_t cand = (wv & 1) ? keep1 : keep0;
          uint32_t word = __shfl(cand, wv >> 1, 32);
          if ((word >> (idx & 31)) & 1u) {      // still kept -> suppress neighbors
            keep0 &= ~m0; keep1 &= ~m1;
          }
        }
      }
    }

    for (int it = 0; it < NWORDS; ++it) {
      uint32_t cand = (it & 1) ? keep1 : keep0;
      uint32_t word = __shfl(cand, it >> 1, 32);
      int n = (it << 5) + lane;
      if (n < NBOX) {
        float f = sc[n];
        float v = (((word >> lane) & 1u) && (f > 0.0f)) ? f : 0.0f;
        prob[((size_t)b * NBOX + n) * NCLS + c] = v;
      }
    }
  }
}

// ----------------------------------------------------------------- launch ---
extern "C" void kernel_launch(void* const* d_in, const int* in_sizes, int n_in,
                              void* d_out, int out_size, void* d_ws, size_t ws_size,
                              hipStream_t stream) {
  (void)in_sizes; (void)n_in; (void)out_size; (void)ws_size;
  const float* x  = (const float*)d_in[0];
  const float* im = (const float*)d_in[1];

  float* prob  = (float*)d_out;
  float* boxes = prob + (size_t)BATCH * NBOX * NCLS;

  uint8_t*  ws      = (uint8_t*)d_ws;
  uint32_t* bits    = (uint32_t*)ws;
  size_t    bitsB   = (size_t)BATCH * NBOX * WPAD * sizeof(uint32_t);
  float*    scores  = (float*)(ws + ((bitsB + 255) & ~(size_t)255));

  decode_kernel <<<dim3(8, BATCH), 256, 0, stream>>>(x, im, boxes, scores);
  ioubits_kernel<<<dim3((NBOX * NWORDS + 255) / 256, BATCH), 256, 0, stream>>>(boxes, bits);
  sortnms_kernel<<<dim3(NCLS, BATCH), 256, 0, stream>>>(scores, bits, prob);
}